// STDLeakyReLu_72842645340569
// MI455X (gfx1250) — compile-verified
//
#include <hip/hip_runtime.h>
#include <hip/hip_bf16.h>
#include <stdint.h>

// ---------------------------------------------------------------------------
// Fused 10-iteration depthwise-3x3 Gaussian stencil for MI455X (gfx1250).
//   u_{k+1} = leaky(o - lam * s * (V * (1-2u_k) * H)),  V=H tridiag(w0,1,w0)
//
//  - One workgroup (256 thr = 8 wave32) owns a 64x64 output tile (halo 16).
//  - o tile: HBM -> LDS once via global_load_async_to_lds_b32 (+s_wait_asynccnt).
//  - Separable conv per 16x16 tile via v_wmma_f32_16x16x4_f32 (8/tile/iter) on
//    the matrix pipe (co-executes with VALU); rank-1 band spillover across
//    tiles handled by masked VALU FMAs fed from the LDS rings.
//  - Tile loop is SGPR-uniform (readfirstlane) so WMMA runs with EXEC=all-1s
//    and tile bookkeeping lands in SALU instead of per-lane VALU.
//  - HBM traffic ~= read o once + write u once (~0.6 GB vs ~8 GB unfused).
// ---------------------------------------------------------------------------

#define TILE   64
#define HALO   16
#define REG    96          // TILE + 2*HALO
#define RSTR   100         // o_s row stride (floats)
#define USTR   100         // u/m1 row stride (98 rows incl ring)
#define NITER  10
#define ALPHA  0.2f

typedef float v2f __attribute__((ext_vector_type(2)));
typedef float v8f __attribute__((ext_vector_type(8)));

__device__ __forceinline__ void async_load_f32_to_lds(const float* gaddr, const float* laddr) {
    unsigned lds_off = (unsigned)(uintptr_t)laddr;           // LDS byte offset
    asm volatile("global_load_async_to_lds_b32 %0, %1, off"
                 :: "v"(lds_off), "v"(gaddr)
                 : "memory");
}

__device__ __forceinline__ void wait_asynccnt0() {
#if __has_builtin(__builtin_amdgcn_s_wait_asynccnt)
    __builtin_amdgcn_s_wait_asynccnt(0);
#else
    asm volatile("s_wait_asynccnt 0" ::: "memory");
#endif
}

__global__ __launch_bounds__(256)
void std_leaky_fused_wmma_kernel(const float* __restrict__ o,
                                 const float* __restrict__ nb_sigma,
                                 const float* __restrict__ lam,
                                 float* __restrict__ out) {
    __shared__ float o_s[REG * RSTR];            // 96 x 100
    __shared__ float u_sh[(REG + 2) * USTR];     // 98 x 100 (ring, u buffer)
    __shared__ float m1_s[(REG + 2) * USTR];     // 98 x 100 (ring rows used)

    const int tx = blockIdx.x & 3;
    const int ty = blockIdx.x >> 2;
    const int c  = blockIdx.y;
    const int b  = blockIdx.z;
    const int x0 = tx * TILE - HALO;
    const int y0 = ty * TILE - HALO;

    const float* op = o   + ((size_t)(b * 64 + c) << 16);    // H*W = 65536
    float*       up = out + ((size_t)(b * 64 + c) << 16);

    const float sig    = nb_sigma[c];
    const float inv2s2 = 1.0f / (2.0f * sig * sig);
    const float w0     = __expf(-inv2s2);
    const float s      = 1.0f / (0.2f * 3.14159265358979f * sig * sig);
    const float ls     = lam[0] * s;

    const int tid  = threadIdx.x;
    const int lane = tid & 31;
    // SGPR wave id: makes the tile loops scalar (s_cbranch), keeps EXEC all-1s
    // around WMMA, and moves /6, %6 tile bookkeeping into SALU.
    const int wave = __builtin_amdgcn_readfirstlane(tid >> 5);

    // ---- Phase 1: async global->LDS fill of the o region ------------------
#pragma unroll
    for (int jx = 0; jx < 3; ++jx) {
        const int x = jx * 32 + lane;
        int gx = x0 + x; gx = gx < 0 ? 0 : (gx > 255 ? 255 : gx);
#pragma unroll
        for (int jy = 0; jy < 12; ++jy) {
            const int y = wave * 12 + jy;
            int gy = y0 + y; gy = gy < 0 ? 0 : (gy > 255 ? 255 : gy);
            async_load_f32_to_lds(op + gy * 256 + gx, &o_s[y * RSTR + x]);
        }
    }

    // ---- Overlap with async: init rings (u ring = 0.5 -> t = 0; m1 ring = 0)
    for (int k = tid; k < REG + 2; k += 256) {
        u_sh[0 * USTR + k]           = 0.5f;   // y = -1
        u_sh[(REG + 1) * USTR + k]   = 0.5f;   // y = 96
        u_sh[k * USTR + 0]           = 0.5f;   // x = -1
        u_sh[k * USTR + (REG + 1)]   = 0.5f;   // x = 96
        m1_s[0 * USTR + k]           = 0.0f;   // m1 ring rows (vert corr)
        m1_s[(REG + 1) * USTR + k]   = 0.0f;
    }

    wait_asynccnt0();
    __syncthreads();

    // ---- Phase 2: u0 = leaky(o), masked outside the image ------------------
#pragma unroll
    for (int jx = 0; jx < 3; ++jx) {
        const int x = jx * 32 + lane;
        const bool inx = (unsigned)(x0 + x) < 256u;
#pragma unroll
        for (int jy = 0; jy < 12; ++jy) {
            const int y = wave * 12 + jy;
            const bool in = inx && ((unsigned)(y0 + y) < 256u);
            const float ov = o_s[y * RSTR + x];
            const float u  = ov >= 0.0f ? ov : ALPHA * ov;
            u_sh[(y + 1) * USTR + (x + 1)] = in ? u : 0.5f;
        }
    }
    __syncthreads();

    // ---- Per-lane WMMA constants ------------------------------------------
    const int n  = lane & 15;      // C/D column, A row, B column
    const int hi = lane >> 4;      // half-wave selector
    const int ko = hi * 2;         // K sub-offset within a 4-chunk

    auto band = [&](int i, int j) -> float {   // tridiag(w0, 1, w0)
        int d = i - j; d = d < 0 ? -d : d;
        return d == 0 ? 1.0f : (d == 1 ? w0 : 0.0f);
    };
    // Same registers serve as A-layout of V and B-layout of H (symmetric).
    v2f bb[4];
#pragma unroll
    for (int q = 0; q < 4; ++q) {
        bb[q].x = band(4 * q + ko,     n);
        bb[q].y = band(4 * q + ko + 1, n);
    }
    const float hmaskw = (n == 0 || n == 15) ? w0 : 0.0f;  // horiz spillover
    const int   hx     = (n == 0) ? -1 : 16;               // neighbor column
    const float vtopw  = (hi == 0) ? w0 : 0.0f;            // row 0 spillover
    const float vbotw  = (hi == 1) ? w0 : 0.0f;            // row 15 spillover

    // ---- Phase 3: 10 iterations; per 16x16 tile: M1 = T*H, M2 = V*M1 -------
    for (int it = 0; it < NITER; ++it) {
        // Stage A: horizontal conv via WMMA (reads u, writes m1)
        for (int t = wave; t < 36; t += 8) {               // scalar loop
            const int ty0 = (t / 6) * 16, tx0 = (t % 6) * 16;   // SALU
            v8f cc = {};
#pragma unroll
            for (int q = 0; q < 4; ++q) {
                const int k = 4 * q + ko;
                const float* p = &u_sh[(ty0 + n + 1) * USTR + (tx0 + k + 1)];
                v2f a;                                     // A = T chunk
                a.x = fmaf(-2.0f, p[0], 1.0f);
                a.y = fmaf(-2.0f, p[1], 1.0f);
                cc = __builtin_amdgcn_wmma_f32_16x16x4_f32(
                        false, a, false, bb[q], (short)0, cc, false, false);
            }
            // cross-tile band spillover into columns 0 / 15 (rank-1, masked)
            {
                const float* pc = &u_sh[(ty0 + 1) * USTR + (tx0 + hx + 1)];
#pragma unroll
                for (int r = 0; r < 8; ++r) {
                    const float tv = fmaf(-2.0f, pc[(r + 8 * hi) * USTR], 1.0f);
                    cc[r] = fmaf(hmaskw, tv, cc[r]);
                }
            }
#pragma unroll
            for (int r = 0; r < 8; ++r)                    // store C/D layout
                m1_s[(ty0 + r + 8 * hi + 1) * USTR + (tx0 + n + 1)] = cc[r];
        }
        __syncthreads();

        // Stage B: vertical conv via WMMA + fused pointwise update (in-place u)
        for (int t = wave; t < 36; t += 8) {               // scalar loop
            const int ty0 = (t / 6) * 16, tx0 = (t % 6) * 16;   // SALU
            v8f cc = {};
#pragma unroll
            for (int q = 0; q < 4; ++q) {
                const int k = 4 * q + ko;
                const float* p = &m1_s[(ty0 + k + 1) * USTR + (tx0 + n + 1)];
                v2f bq; bq.x = p[0]; bq.y = p[USTR];       // B = M1 row chunk
                cc = __builtin_amdgcn_wmma_f32_16x16x4_f32(
                        false, bb[q], false, bq, (short)0, cc, false, false);
            }
            // cross-tile band spillover into rows 0 / 15 (masked)
            cc[0] = fmaf(vtopw, m1_s[(ty0 + 0)  * USTR + (tx0 + n + 1)], cc[0]);
            cc[7] = fmaf(vbotw, m1_s[(ty0 + 17) * USTR + (tx0 + n + 1)], cc[7]);

            const bool inx = (unsigned)(x0 + tx0 + n) < 256u;
#pragma unroll
            for (int r = 0; r < 8; ++r) {
                const int y = ty0 + r + 8 * hi;
                const float nv = fmaf(-ls, cc[r], o_s[y * RSTR + (tx0 + n)]);
                const float u  = nv >= 0.0f ? nv : ALPHA * nv;   // leaky
                const bool in  = inx && ((unsigned)(y0 + y) < 256u);
                u_sh[(y + 1) * USTR + (tx0 + n + 1)] = in ? u : 0.5f;
            }
        }
        __syncthreads();
    }

    // ---- Phase 4: write the 64x64 interior to HBM (coalesced) --------------
#pragma unroll
    for (int jx = 0; jx < 2; ++jx) {
        const int xx = jx * 32 + lane;
#pragma unroll
        for (int jy = 0; jy < 8; ++jy) {
            const int yy = wave * 8 + jy;
            up[(ty * TILE + yy) * 256 + (tx * TILE + xx)] =
                u_sh[(yy + HALO + 1) * USTR + (xx + HALO + 1)];
        }
    }
}

extern "C" void kernel_launch(void* const* d_in, const int* in_sizes, int n_in,
                              void* d_out, int out_size, void* d_ws, size_t ws_size,
                              hipStream_t stream) {
    (void)in_sizes; (void)n_in; (void)d_ws; (void)ws_size; (void)out_size;
    const float* o        = (const float*)d_in[0];   // (16,64,256,256) f32
    const float* nb_sigma = (const float*)d_in[1];   // (64,1,1) f32
    const float* lam      = (const float*)d_in[2];   // (1,) f32
    float* out            = (float*)d_out;

    dim3 grid(16 /* 4x4 tiles */, 64 /* C */, 16 /* B */);
    std_leaky_fused_wmma_kernel<<<grid, dim3(256), 0, stream>>>(o, nb_sigma, lam, out);
}